// DilatedAttention_36086315221656
// MI455X (gfx1250) — compile-verified
//
#include <hip/hip_runtime.h>
#include <math.h>

typedef __attribute__((ext_vector_type(2))) float v2f;
typedef __attribute__((ext_vector_type(8))) float v8f;

#define SEG_LEN 2048
#define DMODEL  1024

__device__ __forceinline__ v2f ld2(const float* p) { return *(const v2f*)p; }

// D = A(16x4,f32) * B(4x16,f32) + C(16x16,f32), wave32 WMMA.
__device__ __forceinline__ v8f wmma_f32(v2f a, v2f b, v8f c) {
    return __builtin_amdgcn_wmma_f32_16x16x4_f32(false, a, false, b, (short)0, c, false, false);
}

// One workgroup = 32 query rows (two 16-row WMMA subtiles). Each wave computes
// both subtiles so every K/V (B-matrix) fragment load feeds 2 WMMAs, halving
// L2 traffic vs a 16-row tile.
__global__ __launch_bounds__(256)
void dilated_attn_wmma(const float* __restrict__ Q,
                       const float* __restrict__ K,
                       const float* __restrict__ V,
                       float* __restrict__ Out,
                       const int r, const int L, const int accum)
{
    // Dynamic LDS: 32*L score/prob buffer + 256 reduce scratch + 32 rowmax + 32 rowinv
    extern __shared__ float smem[];
    float* Sbuf   = smem;             // [32][L]
    float* red    = smem + 32 * L;    // [256]
    float* rowmax = red + 256;        // [32]
    float* rowinv = rowmax + 32;      // [32]

    const int t    = threadIdx.x;
    const int lane = t & 31;
    const int w    = t >> 5;          // wave id 0..7 (wave32)

    const int nqt = L >> 5;                          // 32-row query tiles
    const int qt  = blockIdx.x % nqt;
    const int grp = blockIdx.x / nqt;                // b*4 + seg, 0..7

    const size_t gbase = (size_t)grp * SEG_LEN * DMODEL;
    const float* Qg = Q + gbase;
    const float* Kg = K + gbase;
    const float* Vg = V + gbase;
    float*       Og = Out + gbase;

    // WMMA lane mapping (16x16x4 f32):
    const int m16  = lane & 15;            // A row / B col / C col
    const int kk   = (lane >> 4) << 1;     // K sub-index base: 0 or 2
    const int mrow = (lane >> 4) << 3;     // C row base: 0 or 8
    const size_t rstride = (size_t)r * DMODEL;   // dilated row stride in floats

    // ---------------- Phase 1: Sbuf[32][L] = Q_tile (32xD) * K^T ----------------
    const int ntiles = L >> 4;                       // 16-wide key tiles
    const float* qrow0 = Qg + (size_t)(qt * 32 + m16) * rstride + kk;        // subtile 0
    const float* qrow1 = qrow0 + 16 * rstride;                               // subtile 1
    for (int kt0 = w * 4; kt0 < ntiles; kt0 += 32) { // quads of key tiles per wave
        v8f c0a = {}, c1a = {}, c2a = {}, c3a = {};  // subtile 0 accumulators
        v8f c0b = {}, c1b = {}, c2b = {}, c3b = {};  // subtile 1 accumulators
        const float* k0 = Kg + (size_t)((kt0 + 0) * 16 + m16) * rstride + kk;
        const float* k1 = Kg + (size_t)((kt0 + 1) * 16 + m16) * rstride + kk;
        const float* k2 = Kg + (size_t)((kt0 + 2) * 16 + m16) * rstride + kk;
        const float* k3 = Kg + (size_t)((kt0 + 3) * 16 + m16) * rstride + kk;
        #pragma unroll 2
        for (int dc = 0; dc < DMODEL / 4; ++dc) {
            v2f a0 = ld2(qrow0 + dc * 4);   // A fragments: 2 query subtiles
            v2f a1 = ld2(qrow1 + dc * 4);
            v2f b0 = ld2(k0 + dc * 4);      // B fragments shared by both subtiles
            v2f b1 = ld2(k1 + dc * 4);
            v2f b2 = ld2(k2 + dc * 4);
            v2f b3 = ld2(k3 + dc * 4);
            c0a = wmma_f32(a0, b0, c0a);  c0b = wmma_f32(a1, b0, c0b);
            c1a = wmma_f32(a0, b1, c1a);  c1b = wmma_f32(a1, b1, c1b);
            c2a = wmma_f32(a0, b2, c2a);  c2b = wmma_f32(a1, b2, c2b);
            c3a = wmma_f32(a0, b3, c3a);  c3b = wmma_f32(a1, b3, c3b);
        }
        #pragma unroll
        for (int i = 0; i < 8; ++i) {
            const int ra = (mrow + i) * L;            // subtile 0 rows 0..15
            const int rb = (16 + mrow + i) * L;       // subtile 1 rows 16..31
            Sbuf[ra + (kt0 + 0) * 16 + m16] = c0a[i];
            Sbuf[ra + (kt0 + 1) * 16 + m16] = c1a[i];
            Sbuf[ra + (kt0 + 2) * 16 + m16] = c2a[i];
            Sbuf[ra + (kt0 + 3) * 16 + m16] = c3a[i];
            Sbuf[rb + (kt0 + 0) * 16 + m16] = c0b[i];
            Sbuf[rb + (kt0 + 1) * 16 + m16] = c1b[i];
            Sbuf[rb + (kt0 + 2) * 16 + m16] = c2b[i];
            Sbuf[rb + (kt0 + 3) * 16 + m16] = c3b[i];
        }
    }
    __syncthreads();

    // ---------------- Phase 2: row softmax over 32 rows (8 threads/row) ----------------
    {
        const int row   = t >> 3;      // 0..31
        const int chunk = t & 7;       // 0..7
        const int clen  = L >> 3;      // >= 32
        float* Srow = Sbuf + row * L + chunk * clen;

        float mx = -INFINITY;
        for (int i = 0; i < clen; ++i) mx = fmaxf(mx, Srow[i]);
        red[t] = mx;
        __syncthreads();
        if (chunk == 0) {
            float mm = -INFINITY;
            #pragma unroll
            for (int i = 0; i < 8; ++i) mm = fmaxf(mm, red[row * 8 + i]);
            rowmax[row] = mm;
        }
        __syncthreads();
        const float rm = rowmax[row];
        float sum = 0.f;
        for (int i = 0; i < clen; ++i) {
            const float p = __expf(Srow[i] - rm);
            Srow[i] = p;
            sum += p;
        }
        red[t] = sum;
        __syncthreads();
        if (chunk == 0) {
            float ss = 0.f;
            #pragma unroll
            for (int i = 0; i < 8; ++i) ss += red[row * 8 + i];
            rowinv[row] = 1.0f / ss;
        }
        __syncthreads();
    }

    // ---------------- Phase 3: O(32 x D) = P(32xL) * V(LxD), 16-wide D tiles ----------------
    const int nkc = L >> 2;                       // K-chunks of 4 keys
    for (int dt0 = w * 4; dt0 < DMODEL / 16; dt0 += 32) {  // quads of D tiles per wave
        v8f c0a = {}, c1a = {}, c2a = {}, c3a = {};
        v8f c0b = {}, c1b = {}, c2b = {}, c3b = {};
        const int d0 = (dt0 + 0) * 16 + m16;
        const int d1 = (dt0 + 1) * 16 + m16;
        const int d2 = (dt0 + 2) * 16 + m16;
        const int d3 = (dt0 + 3) * 16 + m16;
        for (int kc = 0; kc < nkc; ++kc) {
            v2f a0 = ld2(&Sbuf[m16 * L + kc * 4 + kk]);         // P fragment subtile 0
            v2f a1 = ld2(&Sbuf[(16 + m16) * L + kc * 4 + kk]);  // P fragment subtile 1
            const float* vr0 = Vg + (size_t)(kc * 4 + kk) * rstride;
            const float* vr1 = vr0 + rstride;
            v2f b0; b0.x = vr0[d0]; b0.y = vr1[d0];  // V fragments shared by both subtiles
            v2f b1; b1.x = vr0[d1]; b1.y = vr1[d1];
            v2f b2; b2.x = vr0[d2]; b2.y = vr1[d2];
            v2f b3; b3.x = vr0[d3]; b3.y = vr1[d3];
            c0a = wmma_f32(a0, b0, c0a);  c0b = wmma_f32(a1, b0, c0b);
            c1a = wmma_f32(a0, b1, c1a);  c1b = wmma_f32(a1, b1, c1b);
            c2a = wmma_f32(a0, b2, c2a);  c2b = wmma_f32(a1, b2, c2b);
            c3a = wmma_f32(a0, b3, c3a);  c3b = wmma_f32(a1, b3, c3b);
        }
        #pragma unroll
        for (int i = 0; i < 8; ++i) {
            const int ma = mrow + i;            // subtile 0 row
            const int mb = 16 + mrow + i;       // subtile 1 row
            const float inva = rowinv[ma];
            const float invb = rowinv[mb];
            float* orowa = Og + (size_t)(qt * 32 + ma) * rstride;
            float* orowb = Og + (size_t)(qt * 32 + mb) * rstride;
            if (accum) {
                orowa[d0] += c0a[i] * inva;  orowb[d0] += c0b[i] * invb;
                orowa[d1] += c1a[i] * inva;  orowb[d1] += c1b[i] * invb;
                orowa[d2] += c2a[i] * inva;  orowb[d2] += c2b[i] * invb;
                orowa[d3] += c3a[i] * inva;  orowb[d3] += c3b[i] * invb;
            } else {
                orowa[d0] = c0a[i] * inva;   orowb[d0] = c0b[i] * invb;
                orowa[d1] = c1a[i] * inva;   orowb[d1] = c1b[i] * invb;
                orowa[d2] = c2a[i] * inva;   orowb[d2] = c2b[i] * invb;
                orowa[d3] = c3a[i] * inva;   orowb[d3] = c3b[i] * invb;
            }
        }
    }
}

extern "C" void kernel_launch(void* const* d_in, const int* in_sizes, int n_in,
                              void* d_out, int out_size, void* d_ws, size_t ws_size,
                              hipStream_t stream) {
    const float* Q = (const float*)d_in[0];
    const float* K = (const float*)d_in[1];
    const float* V = (const float*)d_in[2];
    float* Out = (float*)d_out;

    const int rates[4] = {1, 2, 4, 8};
    for (int i = 0; i < 4; ++i) {
        const int r   = rates[i];
        const int L   = SEG_LEN / r;
        const int nqt = L / 32;                   // 32-row query tiles
        const dim3 grid(8 * nqt);                 // (B*n_seg) * query tiles
        const size_t shmem = (size_t)(32 * L + 256 + 64) * sizeof(float);
        // r=1 covers every position -> plain store; r=2/4/8 accumulate (stream-ordered)
        dilated_attn_wmma<<<grid, dim3(256), shmem, stream>>>(Q, K, V, Out, r, L, i == 0 ? 0 : 1);
    }
}